// AgeGAT_65163243815013
// MI455X (gfx1250) — compile-verified
//
#include <hip/hip_runtime.h>
#include <hip/hip_bf16.h>
#include <math.h>

// ---------------------------------------------------------------------------
// 2-layer GAT + mean-pool + linear, fp32 end-to-end.
// Dense projections use V_WMMA_F32_16X16X4_F32 (fp32 WMMA, 16x16 tiles, K=4).
// Edge softmax/aggregation uses f32 global atomics; gather working set (51MB)
// fits in MI455X's 192MB L2, so the ~2.5GB of edge traffic is L2-resident.
// This revision removes divergent store guards / predicated loads around the
// WMMA core: full 16-row blocks take a uniform fast path with immediate-offset
// stores, and K-padding uses branchless value selects instead of exec masking.
// ---------------------------------------------------------------------------

typedef __attribute__((ext_vector_type(2))) float v2f;
typedef __attribute__((ext_vector_type(8))) float v8f;

#define HIDc 32
#define H1c  4
#define F1c  128          // H1*HID
#define NEG_SLOPE 0.2f

__device__ __forceinline__ float leakyf(float v) { return v > 0.f ? v : NEG_SLOPE * v; }
__device__ __forceinline__ float eluf(float v)   { return v > 0.f ? v : (expf(v) - 1.f); }

// float atomic max via sign-aware integer atomics (portable, no CAS loop)
__device__ __forceinline__ void atomicMaxFloat(float* addr, float v) {
  if (v >= 0.f) atomicMax((int*)addr, __float_as_int(v));
  else          atomicMin((unsigned int*)addr, __float_as_uint(v));
}

// ---------------------------------------------------------------------------
// fill kernel (workspace init each launch; graph-replay safe)
// ---------------------------------------------------------------------------
__global__ void fill_kernel(float* __restrict__ p, float v, long long n) {
  long long i = blockIdx.x * (long long)blockDim.x + threadIdx.x;
  long long stride = (long long)gridDim.x * blockDim.x;
  for (; i < n; i += stride) p[i] = v;
}

// ---------------------------------------------------------------------------
// GEMM1: h1[N,128] = x[N,5] @ W1[5,128] via WMMA f32 16x16x4 (K padded 5->8)
// One wave per 16-row block; 8 N-tiles, 2 k-steps each.
// A layout (16x4 f32): lanes 0-15 -> K=0,1 ; lanes 16-31 -> K=2,3
// B layout (4x16 f32): vgpr0 = rows K=0(l0-15)/K=2(l16-31); vgpr1 = K=1/K=3
// ---------------------------------------------------------------------------
__global__ __launch_bounds__(128) void gemm1_wmma(
    const float* __restrict__ x, const float* __restrict__ W1,
    float* __restrict__ h1, int nRowBlocks, int Ntot) {
  int wave = blockIdx.x * (blockDim.x >> 5) + (threadIdx.x >> 5);
  if (wave >= nRowBlocks) return;                 // uniform per wave
  int lane = threadIdx.x & 31;
  int r    = lane & 15;
  int hi   = lane >> 4;                           // 0: lanes 0-15, 1: lanes 16-31
  float hm = (hi == 0) ? 1.f : 0.f;               // branchless K-padding select
  int row0 = wave * 16;
  bool fullBlock = (row0 + 16 <= Ntot);           // wave-uniform
  int rA   = row0 + r; if (rA >= Ntot) rA = Ntot - 1;   // clamp A loads
  const float* xr = x + (long long)rA * 5;

  // k-step 0: K = 0..3 ;  k-step 1: K = 4 (+ zero padding for K=5,6,7)
  v2f a0; a0.x = xr[2 * hi]; a0.y = xr[2 * hi + 1];     // unconditional loads
  v2f a1; a1.x = xr[4] * hm; a1.y = 0.f;                // value-masked, no exec branch

#pragma unroll
  for (int t = 0; t < 8; ++t) {                   // 8 column tiles of 16
    int col = t * 16 + r;
    v2f b0; b0.x = W1[(2 * hi + 0) * F1c + col];
            b0.y = W1[(2 * hi + 1) * F1c + col];
    v2f b1; b1.x = W1[4 * F1c + col] * hm; b1.y = 0.f;
    v8f c = {};
    c = __builtin_amdgcn_wmma_f32_16x16x4_f32(false, a0, false, b0, (short)0, c, false, false);
    c = __builtin_amdgcn_wmma_f32_16x16x4_f32(false, a1, false, b1, (short)0, c, false, false);
    // C layout: vgpr i -> row i (lanes 0-15) / row i+8 (lanes 16-31)
    if (fullBlock) {
      // single per-lane base + 8 stores with immediate offsets (i*512B)
      float* hp = h1 + (long long)(row0 + hi * 8) * F1c + t * 16 + r;
#pragma unroll
      for (int i = 0; i < 8; ++i) hp[(long long)i * F1c] = c[i];
    } else {
#pragma unroll
      for (int i = 0; i < 8; ++i) {
        int row = row0 + i + hi * 8;
        if (row < Ntot) h1[(long long)row * F1c + t * 16 + r] = c[i];
      }
    }
  }
}

// ---------------------------------------------------------------------------
// GEMM2: h2[N,32] = h1e[N,128] @ W2[128,32] via WMMA f32 16x16x4, 32 k-steps
// ---------------------------------------------------------------------------
__global__ __launch_bounds__(128) void gemm2_wmma(
    const float* __restrict__ h1e, const float* __restrict__ W2,
    float* __restrict__ h2, int nRowBlocks, int Ntot) {
  int wave = blockIdx.x * (blockDim.x >> 5) + (threadIdx.x >> 5);
  if (wave >= nRowBlocks) return;
  int lane = threadIdx.x & 31;
  int r    = lane & 15;
  int hi   = lane >> 4;
  int row0 = wave * 16;
  bool fullBlock = (row0 + 16 <= Ntot);
  int rA   = row0 + r; if (rA >= Ntot) rA = Ntot - 1;
  // A row pointer; offset (4k + 2hi) is even and rows are 512B -> 8B aligned
  const float* ar = h1e + (long long)rA * F1c + 2 * hi;

  v8f c0 = {}, c1 = {};
  for (int k = 0; k < 32; ++k) {                  // K = 128 / 4
    v2f a = *(const v2f*)(ar + 4 * k);            // global_load_b64
    const float* wk = W2 + (4 * k + 2 * hi) * HIDc;
    v2f b0; b0.x = wk[r];      b0.y = wk[HIDc + r];
    v2f b1; b1.x = wk[16 + r]; b1.y = wk[HIDc + 16 + r];
    c0 = __builtin_amdgcn_wmma_f32_16x16x4_f32(false, a, false, b0, (short)0, c0, false, false);
    c1 = __builtin_amdgcn_wmma_f32_16x16x4_f32(false, a, false, b1, (short)0, c1, false, false);
  }
  if (fullBlock) {
    float* hp = h2 + (long long)(row0 + hi * 8) * HIDc + r;
#pragma unroll
    for (int i = 0; i < 8; ++i) {                 // immediate offsets (i*128B, +64B)
      hp[(long long)i * HIDc]      = c0[i];
      hp[(long long)i * HIDc + 16] = c1[i];
    }
  } else {
#pragma unroll
    for (int i = 0; i < 8; ++i) {
      int row = row0 + i + hi * 8;
      if (row < Ntot) {
        h2[(long long)row * HIDc + r]      = c0[i];
        h2[(long long)row * HIDc + 16 + r] = c1[i];
      }
    }
  }
}

// ---------------------------------------------------------------------------
// per-node attention scores: a_s[n,h] = <h[n,h,:], att_src[h,:]> (and dst)
// ---------------------------------------------------------------------------
__global__ void scores1_kernel(const float* __restrict__ h1,
                               const float* __restrict__ att_s,
                               const float* __restrict__ att_d,
                               float* __restrict__ as1, float* __restrict__ ad1, int n) {
  int t = blockIdx.x * blockDim.x + threadIdx.x;
  if (t >= n * H1c) return;
  int node = t >> 2, h = t & 3;
  const float* hp = h1 + (long long)node * F1c + h * HIDc;
  const float* sp = att_s + h * HIDc;
  const float* dp = att_d + h * HIDc;
  float s = 0.f, d = 0.f;
#pragma unroll
  for (int c = 0; c < HIDc; ++c) { float v = hp[c]; s += v * sp[c]; d += v * dp[c]; }
  as1[t] = s; ad1[t] = d;
}

__global__ void scores2_kernel(const float* __restrict__ h2,
                               const float* __restrict__ att_s,
                               const float* __restrict__ att_d,
                               float* __restrict__ as2, float* __restrict__ ad2, int n) {
  int t = blockIdx.x * blockDim.x + threadIdx.x;
  if (t >= n) return;
  const float* hp = h2 + (long long)t * HIDc;
  float s = 0.f, d = 0.f;
#pragma unroll
  for (int c = 0; c < HIDc; ++c) { float v = hp[c]; s += v * att_s[c]; d += v * att_d[c]; }
  as2[t] = s; ad2[t] = d;
}

// ---------------------------------------------------------------------------
// edge passes (edges [0,E) from edge_index, [E,E+n) are self loops)
// ---------------------------------------------------------------------------
template <int H>
__global__ void edge_max_kernel(const int* __restrict__ ei, int E, int n,
                                const float* __restrict__ as, const float* __restrict__ ad,
                                float* __restrict__ m) {
  long long t = blockIdx.x * (long long)blockDim.x + threadIdx.x;
  long long total = (long long)(E + n) * H;
  if (t >= total) return;
  int e = (int)(t / H), h = (int)(t % H);
  int s, d;
  if (e < E) { s = ei[e]; d = ei[E + e]; } else { s = d = e - E; }
  float v = leakyf(as[(long long)s * H + h] + ad[(long long)d * H + h]);
  atomicMaxFloat(&m[(long long)d * H + h], v);
}

template <int H>
__global__ void edge_den_kernel(const int* __restrict__ ei, int E, int n,
                                const float* __restrict__ as, const float* __restrict__ ad,
                                const float* __restrict__ m, float* __restrict__ den) {
  long long t = blockIdx.x * (long long)blockDim.x + threadIdx.x;
  long long total = (long long)(E + n) * H;
  if (t >= total) return;
  int e = (int)(t / H), h = (int)(t % H);
  int s, d;
  if (e < E) { s = ei[e]; d = ei[E + e]; } else { s = d = e - E; }
  float v = leakyf(as[(long long)s * H + h] + ad[(long long)d * H + h]);
  atomicAdd(&den[(long long)d * H + h], expf(v - m[(long long)d * H + h]));
}

// one wave per edge, lane = channel; src/dst uniform per wave -> scalar loads
template <int H>
__global__ void edge_agg_kernel(const int* __restrict__ ei, int E, int n,
                                const float* __restrict__ as, const float* __restrict__ ad,
                                const float* __restrict__ m, const float* __restrict__ den,
                                const float* __restrict__ feat, float* __restrict__ out) {
  int wave = blockIdx.x * (blockDim.x >> 5) + (threadIdx.x >> 5);
  int lane = threadIdx.x & 31;
  int Etot = E + n;
  if (wave >= Etot) return;
  int s, d;
  if (wave < E) { s = ei[wave]; d = ei[E + wave]; } else { s = d = wave - E; }
#pragma unroll
  for (int h = 0; h < H; ++h) {
    float v = leakyf(as[(long long)s * H + h] + ad[(long long)d * H + h]);
    float alpha = expf(v - m[(long long)d * H + h]) / (den[(long long)d * H + h] + 1e-16f);
    float val = alpha * feat[((long long)s * H + h) * HIDc + lane];
    atomicAdd(&out[((long long)d * H + h) * HIDc + lane], val);
  }
}

// ---------------------------------------------------------------------------
// bias + ELU (in place), layer-1 output becomes layer-2 input
// ---------------------------------------------------------------------------
__global__ void bias_elu_kernel(float* __restrict__ buf, const float* __restrict__ bias,
                                int period, long long total) {
  long long i = blockIdx.x * (long long)blockDim.x + threadIdx.x;
  long long stride = (long long)gridDim.x * blockDim.x;
  for (; i < total; i += stride) {
    float v = buf[i] + bias[(int)(i % period)];
    buf[i] = eluf(v);
  }
}

// layer-2 bias + ELU + global mean pool accumulation: one wave per node
__global__ void pool_kernel(const float* __restrict__ out2, const float* __restrict__ b2,
                            const int* __restrict__ batch, float* __restrict__ sums,
                            float* __restrict__ cnt, int n) {
  int wave = blockIdx.x * (blockDim.x >> 5) + (threadIdx.x >> 5);
  int lane = threadIdx.x & 31;
  if (wave >= n) return;
  float v = eluf(out2[(long long)wave * HIDc + lane] + b2[lane]);
  int g = batch[wave];
  atomicAdd(&sums[g * HIDc + lane], v);
  if (lane == 0) atomicAdd(&cnt[g], 1.f);
}

// final 32->1 linear per graph: one wave per graph, wave32 shuffle reduction
__global__ void final_linear_kernel(const float* __restrict__ sums, const float* __restrict__ cnt,
                                    const float* __restrict__ Wlin, const float* __restrict__ blin,
                                    float* __restrict__ out, int G) {
  int g = blockIdx.x;
  int lane = threadIdx.x;
  if (g >= G) return;
  float c = cnt[g]; c = c > 1.f ? c : 1.f;
  float v = (sums[g * HIDc + lane] / c) * Wlin[lane];
#pragma unroll
  for (int o = 16; o > 0; o >>= 1) v += __shfl_xor(v, o, 32);
  if (lane == 0) out[g] = v + blin[0];
}

// ---------------------------------------------------------------------------
extern "C" void kernel_launch(void* const* d_in, const int* in_sizes, int n_in,
                              void* d_out, int out_size, void* d_ws, size_t ws_size,
                              hipStream_t stream) {
  const float* x     = (const float*)d_in[0];
  // d_in[1] = edge_weight : ignored by GATConv (edge_dim=None)
  const float* W1    = (const float*)d_in[2];
  const float* asrc1 = (const float*)d_in[3];
  const float* adst1 = (const float*)d_in[4];
  const float* b1    = (const float*)d_in[5];
  const float* W2    = (const float*)d_in[6];
  const float* asrc2 = (const float*)d_in[7];
  const float* adst2 = (const float*)d_in[8];
  const float* b2    = (const float*)d_in[9];
  const float* Wlin  = (const float*)d_in[10];
  const float* blin  = (const float*)d_in[11];
  const int*   ei    = (const int*)d_in[12];   // [2,E] row-major: src then dst
  const int*   batch = (const int*)d_in[13];

  const int N = in_sizes[13];
  const int E = in_sizes[1];
  const int G = out_size;
  const int Etot = E + N;

  // ---- workspace layout (floats), ~136 MB total ----
  float* ws = (float*)d_ws;
  size_t o = 0;
  float* h1   = ws + o; o += (size_t)N * F1c;    // x @ W1
  float* out1 = ws + o; o += (size_t)N * F1c;    // layer-1 aggregation -> h1e (in place)
  float* as1  = ws + o; o += (size_t)N * H1c;
  float* ad1  = ws + o; o += (size_t)N * H1c;
  float* m1   = ws + o; o += (size_t)N * H1c;
  float* den1 = ws + o; o += (size_t)N * H1c;
  float* h2   = ws + o; o += (size_t)N * HIDc;   // h1e @ W2
  float* out2 = ws + o; o += (size_t)N * HIDc;   // layer-2 aggregation
  float* as2  = ws + o; o += (size_t)N;
  float* ad2  = ws + o; o += (size_t)N;
  float* m2   = ws + o; o += (size_t)N;
  float* den2 = ws + o; o += (size_t)N;
  float* sums = ws + o; o += (size_t)G * HIDc;
  float* cnt  = ws + o; o += (size_t)G;
  (void)ws_size;

  // ---- init accumulators (every launch; deterministic under graph replay) ----
  fill_kernel<<<1024, 256, 0, stream>>>(out1, 0.f, (long long)N * F1c);
  fill_kernel<<<256,  256, 0, stream>>>(den1, 0.f, (long long)N * H1c);
  fill_kernel<<<256,  256, 0, stream>>>(m1, -3.0e38f, (long long)N * H1c);
  fill_kernel<<<512,  256, 0, stream>>>(out2, 0.f, (long long)N * HIDc);
  fill_kernel<<<128,  256, 0, stream>>>(den2, 0.f, (long long)N);
  fill_kernel<<<128,  256, 0, stream>>>(m2, -3.0e38f, (long long)N);
  fill_kernel<<<32,   256, 0, stream>>>(sums, 0.f, (long long)G * HIDc);
  fill_kernel<<<4,    256, 0, stream>>>(cnt, 0.f, (long long)G);

  const int RB = (N + 15) / 16;                   // 16-row WMMA blocks (6250)
  const int wavesPerBlk = 4;                      // 128 threads = 4 waves

  // ---- layer 1 ----
  gemm1_wmma<<<(RB + wavesPerBlk - 1) / wavesPerBlk, 32 * wavesPerBlk, 0, stream>>>(
      x, W1, h1, RB, N);
  scores1_kernel<<<(N * H1c + 255) / 256, 256, 0, stream>>>(h1, asrc1, adst1, as1, ad1, N);
  {
    long long tot = (long long)Etot * H1c;
    int blks = (int)((tot + 255) / 256);
    edge_max_kernel<H1c><<<blks, 256, 0, stream>>>(ei, E, N, as1, ad1, m1);
    edge_den_kernel<H1c><<<blks, 256, 0, stream>>>(ei, E, N, as1, ad1, m1, den1);
  }
  edge_agg_kernel<H1c><<<(Etot + 7) / 8, 256, 0, stream>>>(
      ei, E, N, as1, ad1, m1, den1, h1, out1);
  bias_elu_kernel<<<1024, 256, 0, stream>>>(out1, b1, F1c, (long long)N * F1c);

  // ---- layer 2 ----
  gemm2_wmma<<<(RB + wavesPerBlk - 1) / wavesPerBlk, 32 * wavesPerBlk, 0, stream>>>(
      out1, W2, h2, RB, N);
  scores2_kernel<<<(N + 255) / 256, 256, 0, stream>>>(h2, asrc2, adst2, as2, ad2, N);
  {
    int blks = (Etot + 255) / 256;
    edge_max_kernel<1><<<blks, 256, 0, stream>>>(ei, E, N, as2, ad2, m2);
    edge_den_kernel<1><<<blks, 256, 0, stream>>>(ei, E, N, as2, ad2, m2, den2);
  }
  edge_agg_kernel<1><<<(Etot + 7) / 8, 256, 0, stream>>>(
      ei, E, N, as2, ad2, m2, den2, h2, out2);

  // ---- pool + head ----
  pool_kernel<<<(N + 7) / 8, 256, 0, stream>>>(out2, b2, batch, sums, cnt, N);
  final_linear_kernel<<<G, 32, 0, stream>>>(sums, cnt, Wlin, blin, (float*)d_out, G);
}